// SeqVectorQuantizer_55602646614563
// MI455X (gfx1250) — compile-verified
//
#include <hip/hip_runtime.h>
#include <hip/hip_bf16.h>

#define DIMC   256
#define NSUB   16
#define KCODES 1024
#define EDIM   16
#define DEPTH  3
#define BATCH  8192
#define EMB_STRIDE 18   // padded LDS row stride (matches TDM pad: 2 dwords per 16)

typedef __attribute__((ext_vector_type(2))) float v2f;
typedef __attribute__((ext_vector_type(4))) float v4f;
typedef __attribute__((ext_vector_type(8))) float v8f;
typedef __attribute__((ext_vector_type(4))) unsigned int uivec4;
typedef __attribute__((ext_vector_type(4))) int ivec4;
typedef __attribute__((ext_vector_type(8))) int ivec8;

#if defined(__has_builtin)
#if __has_builtin(__builtin_amdgcn_tensor_load_to_lds) && \
    __has_builtin(__builtin_amdgcn_s_wait_tensorcnt)
#define HAVE_TDM 1
#endif
#endif

// ---- ds_swizzle helpers (group-of-32 xor mode, imm pattern) ----------------
template <int PAT>
__device__ __forceinline__ float swzf(float x) {
    return __int_as_float(__builtin_amdgcn_ds_swizzle(__float_as_int(x), PAT));
}
template <int PAT>
__device__ __forceinline__ int swzi(int x) {
    return __builtin_amdgcn_ds_swizzle(x, PAT);
}
// branchless (val,idx) min step across lanes
template <int PAT>
__device__ __forceinline__ void minstep(float& bv, int& bc) {
    float ov = swzf<PAT>(bv);
    int   oc = swzi<PAT>(bc);
    bool take = (ov < bv) | ((ov == bv) & (oc < bc));
    bv = take ? ov : bv;
    bc = take ? oc : bc;
}
__device__ __forceinline__ float wave_sum(float s) {
    s += swzf<0x401F>(s);   // xor 16
    s += swzf<0x201F>(s);   // xor 8
    s += swzf<0x101F>(s);   // xor 4
    s += swzf<0x081F>(s);   // xor 2
    s += swzf<0x041F>(s);   // xor 1
    return s;
}

// ---------------------------------------------------------------------------
// Kernel 1: residual VQ (3 depths), one sub-codebook per block, 256 batch
// rows, 16 waves each owning a 16-row V_WMMA_F32_16X16X4_F32 tile.
// embedding[n] staged into padded LDS by the Tensor Data Mover.
// ---------------------------------------------------------------------------
__global__ __launch_bounds__(512)
void vq_kernel(const float* __restrict__ z, const float* __restrict__ emb,
               float* __restrict__ qsum_out, float* __restrict__ idxf_out,
               int* __restrict__ idxi_ws)
{
    __shared__ float lds_emb[KCODES * EMB_STRIDE];   // 72 KB (TDM-padded rows)
    __shared__ float lds_esq[KCODES];                //  4 KB
    __shared__ int   lds_idx[16 * 16];               //  1 KB

    const int n       = blockIdx.y;
    const int rowBase = blockIdx.x * 256;
    const int tid     = threadIdx.x;
    const int wave    = tid >> 5;
    const int lane    = tid & 31;
    const int halfHi  = lane >> 4;
    const int n0      = lane & 15;
    const int klo     = halfHi * 2;

    const float* embn = emb + (size_t)n * KCODES * EDIM;

#ifdef HAVE_TDM
    // ---- Tensor Data Mover: 2D tile 16 x 1024 f32, LDS rows padded 16->18 ----
    if (wave == 0) {
        unsigned long long ga = (unsigned long long)(uintptr_t)embn;
        unsigned int lds_base = (unsigned int)(uintptr_t)&lds_emb[0];
        uivec4 g0;
        g0.x = 1u;                                   // count=1 (valid user D#)
        g0.y = lds_base;                             // lds_addr
        g0.z = (unsigned int)ga;                     // global_addr[31:0]
        g0.w = (unsigned int)(ga >> 32) | (2u << 30);// global_addr[56:32] | type=2
        ivec8 g1;
        g1[0] = (2 << 16)      // data_size = 4B
              | (1 << 20)      // pad_enable
              | (3 << 22)      // pad_interval: 16 dwords
              | (1 << 25);     // pad_amount: 2 dwords
        g1[1] = (EDIM & 0xFFFF) << 16;               // tensor_dim0 = 16
        g1[2] = (KCODES & 0xFFFF) << 16;             // tensor_dim1 = 1024
        g1[3] = (EDIM & 0xFFFF) << 16;               // tile_dim0 = 16
        g1[4] = KCODES;                              // tile_dim1 = 1024
        g1[5] = EDIM;                                // tensor_dim0_stride = 16
        g1[6] = 0;
        g1[7] = 0;
        ivec4 zz = {0, 0, 0, 0};
#if __clang_major__ >= 23
        ivec8 z8 = {0, 0, 0, 0, 0, 0, 0, 0};
        __builtin_amdgcn_tensor_load_to_lds(g0, g1, zz, zz, z8, 0);
#else
        __builtin_amdgcn_tensor_load_to_lds(g0, g1, zz, zz, 0);
#endif
        __builtin_amdgcn_s_wait_tensorcnt(0);
    }
#else
    for (int i = tid; i < KCODES * 4; i += 512) {
        int r  = i >> 2;
        int c4 = (i & 3) * 4;
        v4f v = *(const v4f*)(embn + r * EDIM + c4);
        v2f lo; lo.x = v.x; lo.y = v.y;
        v2f hi; hi.x = v.z; hi.y = v.w;
        *(v2f*)&lds_emb[r * EMB_STRIDE + c4 + 0] = lo;
        *(v2f*)&lds_emb[r * EMB_STRIDE + c4 + 2] = hi;
    }
#endif
    __syncthreads();
    // ---- ||e||^2 per code ----
    for (int r = tid; r < KCODES; r += 512) {
        float s = 0.f;
        #pragma unroll
        for (int j = 0; j < EDIM; ++j) {
            float e = lds_emb[r * EMB_STRIDE + j];
            s += e * e;
        }
        lds_esq[r] = s;
    }
    __syncthreads();

    // ---- zr in A-fragment registers (lane holds row M = n0) ----
    const int myRow = rowBase + wave * 16 + n0;
    const float* zrow = z + (size_t)myRow * DIMC + n * EDIM;
    v2f a[4];
    #pragma unroll
    for (int c = 0; c < 4; ++c)
        a[c] = *(const v2f*)(zrow + 4 * c + klo);

    float q_acc[8] = {0.f, 0.f, 0.f, 0.f, 0.f, 0.f, 0.f, 0.f};
    int lastIdx = 0;

    for (int depth = 0; depth < DEPTH; ++depth) {
        float bestVal[8];
        int   bestCode[8];
        #pragma unroll
        for (int v = 0; v < 8; ++v) { bestVal[v] = 3.4e38f; bestCode[v] = 0; }

        for (int t = 0; t < KCODES / 16; ++t) {
            v8f acc = {0.f, 0.f, 0.f, 0.f, 0.f, 0.f, 0.f, 0.f};
            const int coderow = t * 16 + n0;
            #pragma unroll
            for (int c = 0; c < 4; ++c) {
                v2f bf = *(const v2f*)&lds_emb[coderow * EMB_STRIDE + 4 * c + klo];
                acc = __builtin_amdgcn_wmma_f32_16x16x4_f32(
                          false, a[c], false, bf, (short)0, acc, false, false);
            }
            float esq = lds_esq[coderow];
            #pragma unroll
            for (int v = 0; v < 8; ++v) {
                float m = esq - 2.0f * acc[v];
                bool lt = m < bestVal[v];
                bestVal[v]  = lt ? m : bestVal[v];
                bestCode[v] = lt ? coderow : bestCode[v];
            }
        }
        // branchless butterfly min within each 16-lane half
        #pragma unroll
        for (int v = 0; v < 8; ++v) {
            minstep<0x041F>(bestVal[v], bestCode[v]);  // xor 1
            minstep<0x081F>(bestVal[v], bestCode[v]);  // xor 2
            minstep<0x101F>(bestVal[v], bestCode[v]);  // xor 4
            minstep<0x201F>(bestVal[v], bestCode[v]);  // xor 8
        }
        if (n0 == 0) {
            #pragma unroll
            for (int v = 0; v < 8; ++v)
                lds_idx[wave * 16 + halfHi * 8 + v] = bestCode[v];
        }
        int myIdx = lds_idx[wave * 16 + n0];

        #pragma unroll
        for (int c = 0; c < 4; ++c) {
            v2f q = *(const v2f*)&lds_emb[myIdx * EMB_STRIDE + 4 * c + klo];
            a[c] -= q;
            q_acc[2 * c + 0] += q.x;
            q_acc[2 * c + 1] += q.y;
        }
        lastIdx = myIdx;
    }

    float* qrow = qsum_out + (size_t)myRow * DIMC + n * EDIM;
    #pragma unroll
    for (int c = 0; c < 4; ++c) {
        v2f qq; qq.x = q_acc[2 * c]; qq.y = q_acc[2 * c + 1];
        *(v2f*)(qrow + 4 * c + klo) = qq;
    }
    if (halfHi == 0) {
        idxf_out[(size_t)myRow * NSUB + n] = (float)lastIdx;
        idxi_ws [(size_t)myRow * NSUB + n] = lastIdx;
    }
}

// ---------------------------------------------------------------------------
// Kernel 2: z_q = (qsum/3) @ W + b (in-place over qsum region) + loss partials
// ---------------------------------------------------------------------------
#define MSTRIDE 258
__global__ __launch_bounds__(512)
void proj_kernel(const float* __restrict__ z, const float* __restrict__ W,
                 const float* __restrict__ bias, float* __restrict__ zq,
                 float* __restrict__ partials)
{
    __shared__ float lds_m[16 * MSTRIDE];
    __shared__ float lds_red[16];

    const int tid = threadIdx.x, wave = tid >> 5, lane = tid & 31;
    const int n0 = lane & 15, halfHi = lane >> 4, klo = halfHi * 2;
    const int rowBase = blockIdx.x * 16;
    const int colBase = wave * 16;

    const float s = 1.0f / (float)DEPTH;
    for (int i = tid; i < 16 * 64; i += 512) {
        int r = i >> 6, c4 = (i & 63) * 4;
        v4f v = *(const v4f*)(zq + (size_t)(rowBase + r) * DIMC + c4);
        v2f lo; lo.x = v.x * s; lo.y = v.y * s;
        v2f hi; hi.x = v.z * s; hi.y = v.w * s;
        *(v2f*)&lds_m[r * MSTRIDE + c4 + 0] = lo;
        *(v2f*)&lds_m[r * MSTRIDE + c4 + 2] = hi;
    }
    __syncthreads();

    v8f acc = {0.f, 0.f, 0.f, 0.f, 0.f, 0.f, 0.f, 0.f};
    for (int kc = 0; kc < DIMC / 4; ++kc) {
        v2f af = *(const v2f*)&lds_m[n0 * MSTRIDE + 4 * kc + klo];
        v2f bf;
        bf.x = W[(size_t)(4 * kc + klo + 0) * DIMC + colBase + n0];
        bf.y = W[(size_t)(4 * kc + klo + 1) * DIMC + colBase + n0];
        acc = __builtin_amdgcn_wmma_f32_16x16x4_f32(
                  false, af, false, bf, (short)0, acc, false, false);
    }
    float bj = bias[colBase + n0];
    float sumsq = 0.f;
    #pragma unroll
    for (int v = 0; v < 8; ++v) {
        int row = rowBase + halfHi * 8 + v;
        int col = colBase + n0;
        float zqv  = acc[v] + bj;
        float zval = z[(size_t)row * DIMC + col];
        float d = zqv - zval;
        sumsq += d * d;
        zq[(size_t)row * DIMC + col] = zqv;
    }
    sumsq = wave_sum(sumsq);
    if (lane == 0) lds_red[wave] = sumsq;
    __syncthreads();
    if (tid == 0) {
        float t = 0.f;
        #pragma unroll
        for (int wv = 0; wv < 16; ++wv) t += lds_red[wv];
        partials[blockIdx.x] = t;
    }
}

// ---------------------------------------------------------------------------
// Kernel 3: one-hot writer — 537 MB of nontemporal b128 streaming stores
// ---------------------------------------------------------------------------
__global__ __launch_bounds__(256)
void onehot_kernel(const int* __restrict__ idx, float* __restrict__ oh)
{
    const int rowBase = blockIdx.x * 16;
    const int k = threadIdx.x * 4;
    for (int r = 0; r < 16; ++r) {
        const int row = rowBase + r;
        const int id  = idx[row];
        v4f v;
        v.x = (k + 0 == id) ? 1.f : 0.f;
        v.y = (k + 1 == id) ? 1.f : 0.f;
        v.z = (k + 2 == id) ? 1.f : 0.f;
        v.w = (k + 3 == id) ? 1.f : 0.f;
        __builtin_nontemporal_store(v, (v4f*)(oh + (size_t)row * KCODES + k));
    }
}

// ---------------------------------------------------------------------------
// Kernel 4: final loss reduction (deterministic)
// ---------------------------------------------------------------------------
__global__ __launch_bounds__(512)
void loss_kernel(const float* __restrict__ partials, int n, float* __restrict__ out)
{
    __shared__ float red[16];
    float s = 0.f;
    for (int i = threadIdx.x; i < n; i += 512) s += partials[i];
    s = wave_sum(s);
    const int wave = threadIdx.x >> 5, lane = threadIdx.x & 31;
    if (lane == 0) red[wave] = s;
    __syncthreads();
    if (threadIdx.x == 0) {
        float t = 0.f;
        #pragma unroll
        for (int wv = 0; wv < 16; ++wv) t += red[wv];
        out[0] = t * (1.25f / (float)(BATCH * DIMC));
    }
}

// ---------------------------------------------------------------------------
extern "C" void kernel_launch(void* const* d_in, const int* in_sizes, int n_in,
                              void* d_out, int out_size, void* d_ws, size_t ws_size,
                              hipStream_t stream)
{
    (void)in_sizes; (void)n_in; (void)out_size; (void)ws_size;
    const float* z    = (const float*)d_in[0];
    const float* emb  = (const float*)d_in[1];
    const float* W    = (const float*)d_in[2];
    const float* bias = (const float*)d_in[3];

    float* out     = (float*)d_out;
    float* loss    = out;                                      // [1]
    float* zq      = out + 1;                                  // [B*256]
    float* onehot  = zq + (size_t)BATCH * DIMC;                // [B*16*1024]
    float* idxf    = onehot + (size_t)BATCH * NSUB * KCODES;   // [B*16]

    int*   idxi     = (int*)d_ws;                              // B*16 ints
    float* partials = (float*)((char*)d_ws + (size_t)BATCH * NSUB * sizeof(int));

    vq_kernel<<<dim3(BATCH / 256, NSUB), 512, 0, stream>>>(z, emb, zq, idxf, idxi);
    proj_kernel<<<BATCH / 16, 512, 0, stream>>>(z, W, bias, zq, partials);
    onehot_kernel<<<BATCH * NSUB / 16, 256, 0, stream>>>(idxi, onehot);
    loss_kernel<<<1, 512, 0, stream>>>(partials, BATCH / 16, loss);
}